// LightGCN_83442624627044
// MI455X (gfx1250) — compile-verified
//
#include <hip/hip_runtime.h>
#include <stdint.h>

#define EMB 64

static constexpr int kTile    = 1024;            // nnz staged per block tile
static constexpr int kBlock   = 256;             // 8 waves (wave32)
static constexpr int kWaves   = kBlock / 32;     // 8
static constexpr int kPerWave = kTile / kWaves;  // 128 nnz per wave per tile

// ---------------- CDNA5 async global->LDS helpers ----------------
typedef int   v4i __attribute__((vector_size(16)));
typedef float v4f __attribute__((ext_vector_type(4)));   // native vector for NT builtins
typedef __attribute__((address_space(1))) v4i gas_v4i;   // global (AS1)
typedef __attribute__((address_space(3))) v4i las_v4i;   // LDS (AS3)

__device__ __forceinline__ void async_b128(const void* g, void* l) {
#if __has_builtin(__builtin_amdgcn_global_load_async_to_lds_b128)
  __builtin_amdgcn_global_load_async_to_lds_b128(
      (gas_v4i*)(uintptr_t)g, (las_v4i*)(uintptr_t)l, 0, 0);
#else
  unsigned loff = (unsigned)(uintptr_t)l;  // low 32 bits of flat addr = LDS offset
  asm volatile("global_load_async_to_lds_b128 %0, %1, off"
               :: "v"(loff), "v"(g) : "memory");
#endif
}

template <int N>
__device__ __forceinline__ void wait_async() {
#if __has_builtin(__builtin_amdgcn_s_wait_asynccnt)
  __builtin_amdgcn_s_wait_asynccnt(N);
#else
  asm volatile("s_wait_asynccnt %0" :: "i"(N) : "memory");
#endif
}

// ---------------- kernels ----------------
__global__ __launch_bounds__(256)
void lgcn_init_kernel(const float* __restrict__ users, const float* __restrict__ items,
                      float* __restrict__ ego, float* __restrict__ acc,
                      long long uCount, long long total) {
  long long i = ((long long)blockIdx.x * blockDim.x + threadIdx.x) * 4;
  long long stride = (long long)gridDim.x * blockDim.x * 4;
  for (; i < total; i += stride) {
    v4f v;
    if (i < uCount) v = *(const v4f*)(users + i);
    else            v = *(const v4f*)(items + (i - uCount));
    *(v4f*)(ego + i) = v;
    *(v4f*)(acc + i) = v;
  }
}

__global__ __launch_bounds__(256)
void lgcn_zero_kernel(float* __restrict__ p, long long count4) {
  long long i = (long long)blockIdx.x * blockDim.x + threadIdx.x;
  if (i < count4) ((v4f*)p)[i] = (v4f){0.f, 0.f, 0.f, 0.f};
}

__global__ __launch_bounds__(256)
void lgcn_spmm_kernel(const float* __restrict__ x, float* __restrict__ y,
                      const float* __restrict__ vals,
                      const int* __restrict__ rows,
                      const int* __restrict__ cols,
                      int nnz) {
  __shared__ int   sRows[2][kTile];
  __shared__ int   sCols[2][kTile];
  __shared__ float sVals[2][kTile];

  const int tid  = threadIdx.x;
  const int lane = tid & 31;   // wave32
  const int wave = tid >> 5;
  const int e0   = tid * 4;
  const int nTiles = (nnz + kTile - 1) / kTile;

  // Stage tile `t` into LDS buffer `b`. Full tiles use the ASYNCcnt-tracked
  // async DMA path (3 x b128 per thread); the (single, final) partial tile
  // falls back to direct LDS stores (DScnt, flushed by __syncthreads).
  auto stage = [&](int t, int b) -> bool {
    const long long base = (long long)t * kTile;
    if (base + kTile <= (long long)nnz) {
      async_b128(rows + base + e0, &sRows[b][e0]);
      async_b128(cols + base + e0, &sCols[b][e0]);
      async_b128(vals + base + e0, &sVals[b][e0]);
      return true;
    }
    for (int k = 0; k < 4; ++k) {
      long long idx = base + e0 + k;
      if (idx < (long long)nnz) {
        sRows[b][e0 + k] = rows[idx];
        sCols[b][e0 + k] = cols[idx];
        sVals[b][e0 + k] = vals[idx];
      }
    }
    return false;
  };

  int tile = blockIdx.x;
  if (tile >= nTiles) return;

  int buf = 0;
  stage(tile, buf);  // prologue

  for (; tile < nTiles; tile += gridDim.x) {
    const long long base = (long long)tile * kTile;

    // Pipeline: issue next tile's async DMA before waiting on the current one.
    const int nextTile = tile + gridDim.x;
    bool nextAsync = false;
    if (nextTile < nTiles) nextAsync = stage(nextTile, buf ^ 1);

    // Async loads complete in-order per wave: waiting until <=3 remain
    // guarantees the current tile's 3 loads have landed in LDS.
    if (nextAsync) wait_async<3>();
    else           wait_async<0>();
    __syncthreads();

    const int kBeg = wave * kPerWave;
    const int kEnd = kBeg + kPerWave;
    for (int k = kBeg; k < kEnd; ++k) {
      if (base + k >= (long long)nnz) break;
      const int   r = sRows[buf][k];   // LDS broadcast (same addr, all lanes)
      const int   c = sCols[buf][k];
      const float v = sVals[buf][k];
      const float2 e = *((const float2*)(x + (long long)c * EMB) + lane);
      float* dst = y + (long long)r * EMB + lane * 2;
      unsafeAtomicAdd(dst,     v * e.x);   // global_atomic_add_f32, no return
      unsafeAtomicAdd(dst + 1, v * e.y);
    }
    __syncthreads();  // all waves done reading `buf` before it is re-staged
    buf ^= 1;
  }
}

__global__ __launch_bounds__(256)
void lgcn_accum_kernel(float* __restrict__ acc, const float* __restrict__ add, long long count4) {
  long long i = (long long)blockIdx.x * blockDim.x + threadIdx.x;
  if (i < count4) {
    // acc is streaming-only: keep it non-temporal so the L2 stays dedicated
    // to the embedding tables used by the gather/atomic phase.
    v4f a = __builtin_nontemporal_load(((const v4f*)acc) + i);
    v4f b = ((const v4f*)add)[i];   // regular: add becomes next gather src
    a += b;
    __builtin_nontemporal_store(a, ((v4f*)acc) + i);
  }
}

__global__ __launch_bounds__(256)
void lgcn_scale_kernel(float* __restrict__ p, float s, long long count4) {
  long long i = (long long)blockIdx.x * blockDim.x + threadIdx.x;
  if (i < count4) {
    v4f a = __builtin_nontemporal_load(((const v4f*)p) + i);
    a *= s;
    __builtin_nontemporal_store(a, ((v4f*)p) + i);
  }
}

// ---------------- host launcher ----------------
extern "C" void kernel_launch(void* const* d_in, const int* in_sizes, int n_in,
                              void* d_out, int out_size, void* d_ws, size_t ws_size,
                              hipStream_t stream) {
  const float* users = (const float*)d_in[0];
  const float* items = (const float*)d_in[1];
  const float* vals  = (const float*)d_in[2];
  const int*   rows  = (const int*)d_in[3];
  const int*   cols  = (const int*)d_in[4];

  const long long uN  = (long long)in_sizes[0] / EMB;
  const long long iN  = (long long)in_sizes[1] / EMB;
  const long long N   = uN + iN;
  const int nnz       = in_sizes[2];

  float* acc = (float*)d_out;           // accumulator lives in d_out
  float* ego = (float*)d_ws;            // [N*EMB]
  float* nxt = ego + N * EMB;           // [N*EMB]

  const long long total  = N * EMB;
  const long long count4 = total / 4;
  const int tBlocks = (int)((count4 + kBlock - 1) / kBlock);

  lgcn_init_kernel<<<4096, kBlock, 0, stream>>>(users, items, ego, acc, uN * EMB, total);

  const int nTiles = (nnz + kTile - 1) / kTile;
  const int spmmBlocks = nTiles < 2048 ? nTiles : 2048;

  for (int l = 0; l < 3; ++l) {
    lgcn_zero_kernel<<<tBlocks, kBlock, 0, stream>>>(nxt, count4);
    lgcn_spmm_kernel<<<spmmBlocks, kBlock, 0, stream>>>(ego, nxt, vals, rows, cols, nnz);
    lgcn_accum_kernel<<<tBlocks, kBlock, 0, stream>>>(acc, nxt, count4);
    float* t = ego; ego = nxt; nxt = t;
  }
  lgcn_scale_kernel<<<tBlocks, kBlock, 0, stream>>>(acc, 0.25f, count4);
}